// CustomWeightedGNN_72232759984606
// MI455X (gfx1250) — compile-verified
//
#include <hip/hip_runtime.h>

typedef __attribute__((ext_vector_type(2))) float v2f;
typedef __attribute__((ext_vector_type(8))) float v8f;

#define N_NODES 10000
#define N_EDGES 640000
#define D_IN    128
#define D_HID   256
#define D_OUT   64

// ---------------------------------------------------------------- CSR build
__global__ void zero_i32(int* __restrict__ p, int n) {
  int i = blockIdx.x * blockDim.x + threadIdx.x;
  if (i < n) p[i] = 0;
}

__global__ void count_deg(const int* __restrict__ dst, int* __restrict__ deg, int nE) {
  int e = blockIdx.x * blockDim.x + threadIdx.x;
  if (e < nE) atomicAdd(&deg[dst[e]], 1);
}

// Single-block (1024 thr = 32 waves) exclusive scan over N_NODES degrees -> rowptr.
// LDS Hillis-Steele per 1024-chunk with running carry.
__global__ void exclusive_scan(const int* __restrict__ deg, int* __restrict__ rowptr, int n) {
  __shared__ int smem[1024];
  __shared__ int carry;
  int t = threadIdx.x;
  if (t == 0) carry = 0;
  __syncthreads();
  for (int base = 0; base < n; base += 1024) {
    int i = base + t;
    int v = (i < n) ? deg[i] : 0;
    smem[t] = v;
    __syncthreads();
    #pragma unroll
    for (int off = 1; off < 1024; off <<= 1) {
      int x = (t >= off) ? smem[t - off] : 0;
      __syncthreads();
      smem[t] += x;
      __syncthreads();
    }
    int cbase = carry;
    if (i < n) rowptr[i] = cbase + smem[t] - v;   // exclusive
    __syncthreads();
    if (t == 0) carry = cbase + smem[1023];
    __syncthreads();
  }
  if (t == 0) rowptr[n] = carry;                  // == nE
}

__global__ void inv_from_deg(const int* __restrict__ deg, float* __restrict__ inv, int n) {
  int i = blockIdx.x * blockDim.x + threadIdx.x;
  if (i < n) {
    int d = deg[i];
    inv[i] = (d > 0) ? (1.0f / (float)d) : 0.0f;  // where(deg>0, 1/deg, 0)
  }
}

__global__ void copy_i32(const int* __restrict__ a, int* __restrict__ b, int n) {
  int i = blockIdx.x * blockDim.x + threadIdx.x;
  if (i < n) b[i] = a[i];
}

__global__ void bin_edges(const int* __restrict__ src, const int* __restrict__ dst,
                          const float* __restrict__ w,
                          int* __restrict__ cursor,
                          int* __restrict__ es, float* __restrict__ ew, int nE) {
  int e = blockIdx.x * blockDim.x + threadIdx.x;
  if (e < nE) {
    int pos = atomicAdd(&cursor[dst[e]], 1);
    es[pos] = src[e];
    ew[pos] = w[e];
  }
}

// ------------------------------------------------------- per-node CSR gather
// One wave per destination node; lane carries D/32 features in registers.
// Writes mean-scaled result straight into concat buffer column block `coff`.
template<int D>
__global__ void aggregate_nodes(const float* __restrict__ X, int ldx,
                                const int* __restrict__ rowptr,
                                const int* __restrict__ es,
                                const float* __restrict__ ew,
                                const float* __restrict__ inv,
                                float* Y, int ldy, int coff, int nN) {
  int node = (blockIdx.x * blockDim.x + threadIdx.x) >> 5;
  int lane = threadIdx.x & 31;
  if (node >= nN) return;
  int beg = rowptr[node];
  int end = rowptr[node + 1];
  float4 acc[D / 128];
  #pragma unroll
  for (int c = 0; c < D / 128; ++c) acc[c] = make_float4(0.f, 0.f, 0.f, 0.f);
  for (int j = beg; j < end; ++j) {
    int s = es[j];
    float ww = ew[j];
    const float4* xr = (const float4*)(X + (size_t)s * ldx);
    #pragma unroll
    for (int c = 0; c < D / 128; ++c) {
      float4 v = xr[lane + 32 * c];
      acc[c].x += v.x * ww;
      acc[c].y += v.y * ww;
      acc[c].z += v.z * ww;
      acc[c].w += v.w * ww;
    }
  }
  float iv = inv[node];
  float4* yr = (float4*)(Y + (size_t)node * ldy + coff);
  #pragma unroll
  for (int c = 0; c < D / 128; ++c)
    yr[lane + 32 * c] = make_float4(acc[c].x * iv, acc[c].y * iv, acc[c].z * iv, acc[c].w * iv);
}

__global__ void copy_self(const float* __restrict__ h, float* __restrict__ Y, int nN) {
  int i = blockIdx.x * blockDim.x + threadIdx.x;
  if (i >= nN * D_IN) return;
  int n = i >> 7;            // /128
  int f = i & 127;
  Y[(size_t)n * (2 * D_IN) + f] = h[i];
}

// ---------------------------------------------------------------- fp32 WMMA GEMM
// Wave computes 16(M) x 64(N): 4 accumulators, K-step 4 via V_WMMA_F32_16X16X4_F32.
// NDIM/KDIM compile-time => B offsets are immediates, B pointer bumps by constant.
template<int NDIM, int KDIM, bool RELU>
__global__ void wmma_gemm_f32(const float* __restrict__ A, int lda,
                              const float* __restrict__ B,
                              const float* __restrict__ bias,
                              float* __restrict__ Y, int ldy,
                              int Mtiles, int Nstrips) {
  int wid  = (blockIdx.x * blockDim.x + threadIdx.x) >> 5;
  int lane = threadIdx.x & 31;
  if (wid >= Mtiles * Nstrips) return;
  int m0 = (wid / Nstrips) * 16;
  int n0 = (wid % Nstrips) * 64;
  int half = lane >> 4;
  int mr   = lane & 15;

  const float* ap = A + (size_t)(m0 + mr) * lda + 2 * half;
  const float* bp = B + (2 * half) * NDIM + n0 + mr;
  v8f acc0 = {}, acc1 = {}, acc2 = {}, acc3 = {};

  #pragma unroll 2
  for (int k = 0; k < KDIM; k += 4) {
    v2f a = *(const v2f*)ap;
    ap += 4;
    __builtin_prefetch(ap + 60, 0, 3);             // near-cache speculative prefetch
    v2f b;
    b.x = bp[0];         b.y = bp[NDIM];
    acc0 = __builtin_amdgcn_wmma_f32_16x16x4_f32(false, a, false, b, (short)0, acc0, false, false);
    b.x = bp[16];        b.y = bp[NDIM + 16];
    acc1 = __builtin_amdgcn_wmma_f32_16x16x4_f32(false, a, false, b, (short)0, acc1, false, false);
    b.x = bp[32];        b.y = bp[NDIM + 32];
    acc2 = __builtin_amdgcn_wmma_f32_16x16x4_f32(false, a, false, b, (short)0, acc2, false, false);
    b.x = bp[48];        b.y = bp[NDIM + 48];
    acc3 = __builtin_amdgcn_wmma_f32_16x16x4_f32(false, a, false, b, (short)0, acc3, false, false);
    bp += 4 * NDIM;
  }

  #pragma unroll
  for (int j = 0; j < 4; ++j) {
    v8f acc = (j == 0) ? acc0 : (j == 1) ? acc1 : (j == 2) ? acc2 : acc3;
    int n = n0 + 16 * j + mr;
    float bv = bias[n];
    #pragma unroll
    for (int r = 0; r < 8; ++r) {
      int row = m0 + r + 8 * half;
      float v = acc[r] + bv;
      if (RELU) v = (v > 0.0f) ? v : 0.0f;
      Y[(size_t)row * ldy + n] = v;
    }
  }
}

// ---------------------------------------------------------------- launch
extern "C" void kernel_launch(void* const* d_in, const int* in_sizes, int n_in,
                              void* d_out, int out_size, void* d_ws, size_t ws_size,
                              hipStream_t stream) {
  const float* h   = (const float*)d_in[0];
  const float* w   = (const float*)d_in[1];
  const int*   src = (const int*)d_in[2];
  const int*   dst = (const int*)d_in[3];
  const float* W1  = (const float*)d_in[4];
  const float* b1  = (const float*)d_in[5];
  const float* W2  = (const float*)d_in[6];
  const float* b2  = (const float*)d_in[7];
  float* out = (float*)d_out;

  // Workspace layout (32-bit words):
  int*   deg    = (int*)d_ws;                   // 10240
  int*   rowptr = deg + 10240;                  // 10240 (needs 10001)
  int*   cursor = rowptr + 10240;               // 10240
  int*   es     = cursor + 10240;               // 640000 (sorted-by-dst src idx)
  float* ew     = (float*)(es + N_EDGES);       // 640000 (sorted-by-dst weight)
  float* inv    = ew + N_EDGES;                 // 10240
  float* bufA   = inv + 10240;                  // 10000*256  X1 concat
  float* bufB   = bufA + N_NODES * D_HID;       // 10000*512  X2 concat (0:256 = relu(X1))

  // ---- CSR build (shared by both layers) ----
  zero_i32<<<(N_NODES + 255) / 256, 256, 0, stream>>>(deg, N_NODES);
  count_deg<<<(N_EDGES + 255) / 256, 256, 0, stream>>>(dst, deg, N_EDGES);
  exclusive_scan<<<1, 1024, 0, stream>>>(deg, rowptr, N_NODES);
  inv_from_deg<<<(N_NODES + 255) / 256, 256, 0, stream>>>(deg, inv, N_NODES);
  copy_i32<<<(N_NODES + 255) / 256, 256, 0, stream>>>(rowptr, cursor, N_NODES);
  bin_edges<<<(N_EDGES + 255) / 256, 256, 0, stream>>>(src, dst, w, cursor, es, ew, N_EDGES);

  // ---- layer 1 ----
  copy_self<<<(N_NODES * D_IN + 255) / 256, 256, 0, stream>>>(h, bufA, N_NODES);
  aggregate_nodes<D_IN><<<(N_NODES * 32 + 255) / 256, 256, 0, stream>>>(
      h, D_IN, rowptr, es, ew, inv, bufA, 2 * D_IN, D_IN, N_NODES);
  // [10000,256] @ [256,256] + b1, relu -> bufB[:, 0:256] (ldy=512)
  {
    int Mtiles = N_NODES / 16, Nstrips = D_HID / 64;          // 2500 waves
    wmma_gemm_f32<D_HID, 2 * D_IN, true><<<(Mtiles * Nstrips + 7) / 8, 256, 0, stream>>>(
        bufA, 2 * D_IN, W1, b1, bufB, 2 * D_HID, Mtiles, Nstrips);
  }

  // ---- layer 2 ----
  aggregate_nodes<D_HID><<<(N_NODES * 32 + 255) / 256, 256, 0, stream>>>(
      bufB, 2 * D_HID, rowptr, es, ew, inv, bufB, 2 * D_HID, D_HID, N_NODES);
  // [10000,512] @ [512,64] + b2 -> out (ldy=64)
  {
    int Mtiles = N_NODES / 16, Nstrips = D_OUT / 64;          // 625 waves
    wmma_gemm_f32<D_OUT, 2 * D_HID, false><<<(Mtiles * Nstrips + 7) / 8, 256, 0, stream>>>(
        bufB, 2 * D_HID, W2, b2, out, D_OUT, Mtiles, Nstrips);
  }
}